// Class_Rank_79147657331264
// MI455X (gfx1250) — compile-verified
//
#include <hip/hip_runtime.h>

// Rank-N-Contrast loss for 6 classes x 256 samples x 256-dim features on gfx1250.
// Gram matrices computed with V_WMMA_F32_16X16X4_F32 (fp32-exact matrix path).
// All reductions are fixed-order => bitwise deterministic across graph replays.

#define NUM_CLASSES 6
#define GROUP 256
#define DIMF 256
#define LAMW 0.8f

typedef float v2f __attribute__((ext_vector_type(2)));
typedef float v8f __attribute__((ext_vector_type(8)));

// ---------------------------------------------------------------- row norms
// One wave32 per feature row: lane-strided loads (fully coalesced) + shuffle tree.
__global__ void k_norms(const float* __restrict__ F, float* __restrict__ n2) {
    int wave = (blockIdx.x * blockDim.x + threadIdx.x) >> 5;
    int lane = threadIdx.x & 31;
    if (wave >= NUM_CLASSES * GROUP) return;
    const float* row = F + (size_t)wave * DIMF;
    float s = 0.f;
#pragma unroll
    for (int t = 0; t < DIMF / 32; ++t) {
        float v = row[lane + 32 * t];
        s += v * v;
    }
#pragma unroll
    for (int off = 16; off > 0; off >>= 1) s += __shfl_down(s, off, 32);
    if (lane == 0) n2[wave] = s;
}

// ------------------------------------------------------------ label distance
// L[c][i][j] = LAM*rot + (1-LAM)*l1(trans); rot depends on class branch.
__global__ void k_labeldist(const float* __restrict__ g, const float* __restrict__ r,
                            const float* __restrict__ t, const int* __restrict__ sym,
                            float* __restrict__ L) {
    int tid = blockIdx.x * blockDim.x + threadIdx.x;
    if (tid >= NUM_CLASSES * GROUP * GROUP) return;
    int c   = tid >> 16;           // / (256*256)
    int rem = tid & 65535;
    int i   = rem >> 8;
    int j   = rem & 255;
    int gi = c * GROUP + i, gj = c * GROUP + j;

    float l1g = fabsf(g[gi*3+0]-g[gj*3+0]) + fabsf(g[gi*3+1]-g[gj*3+1]) + fabsf(g[gi*3+2]-g[gj*3+2]);
    float l1t = fabsf(t[gi*3+0]-t[gj*3+0]) + fabsf(t[gi*3+1]-t[gj*3+1]) + fabsf(t[gi*3+2]-t[gj*3+2]);

    float rot;
    if (c < 2) {                       // fully symmetric classes: green only
        rot = l1g;
    } else {
        float l1r = fabsf(r[gi*3+0]-r[gj*3+0]) + fabsf(r[gi*3+1]-r[gj*3+1]) + fabsf(r[gi*3+2]-r[gj*3+2]);
        if (c < 5) {                   // non-symmetric classes
            rot = 0.5f * (l1g + l1r);
        } else {                       // mug class: red only when both non-sym
            float bi = (sym[gi*4] == 0) ? 1.f : 0.f;
            float bj = (sym[gj*4] == 0) ? 1.f : 0.f;
            float both = bi * bj;
            rot = (l1g + both * l1r) / (1.f + both);
        }
    }
    L[tid] = LAMW * rot + (1.f - LAMW) * l1t;
}

// --------------------------------------------------- WMMA Gram -> distances
// 8 waves/block, one 16x16 output tile per wave. 1536 tiles total.
// A-layout (16x4 f32): lane l -> M = l&15, K-pair = {2*(l>=16), 2*(l>=16)+1}.
// For G = F*F^T, B is loaded with the identical per-lane pattern on row N.
__global__ void k_gram_dist(const float* __restrict__ F, const float* __restrict__ n2,
                            float* __restrict__ Dist) {
    int wave = threadIdx.x >> 5;
    int lane = threadIdx.x & 31;
    int tile = blockIdx.x * 8 + wave;            // < 6*16*16 = 1536
    int c    = tile >> 8;
    int rem  = tile & 255;
    int tm   = (rem >> 4) << 4;
    int tn   = (rem & 15) << 4;
    int hi   = lane >> 4;                        // K-group select (0 or 1)
    int lm   = lane & 15;

    const float* arow = F + (size_t)(c * GROUP + tm + lm) * DIMF + 2 * hi;
    const float* brow = F + (size_t)(c * GROUP + tn + lm) * DIMF + 2 * hi;

    v8f acc = {0.f, 0.f, 0.f, 0.f, 0.f, 0.f, 0.f, 0.f};
#pragma unroll 8
    for (int kb = 0; kb < DIMF; kb += 4) {
        v2f a = *(const v2f*)(arow + kb);
        v2f b = *(const v2f*)(brow + kb);
        // 8 args: (neg_a, A, neg_b, B, c_mod, C, reuse_a, reuse_b)
        acc = __builtin_amdgcn_wmma_f32_16x16x4_f32(
            false, a, false, b, (short)0, acc, false, false);
    }

    // C/D layout: VGPR rr, lane l -> M = tm + rr + 8*hi, N = tn + (l&15)
    float nb = n2[c * GROUP + tn + lm];
#pragma unroll
    for (int rr = 0; rr < 8; ++rr) {
        int m = tm + rr + 8 * hi;
        float d2 = n2[c * GROUP + m] + nb - 2.f * acc[rr];
        d2 = fmaxf(d2, 0.f) + 1e-12f;
        Dist[((size_t)c * GROUP + m) * GROUP + tn + lm] = sqrtf(d2);
    }
}

// -------------------------------------------------------------- per-row RnC
// One block per (class,row): row-max, exp, masked denominator, log, block sum.
__global__ void k_rows(const float* __restrict__ Dist, const float* __restrict__ L,
                       float* __restrict__ rowsum) {
    __shared__ float sE[GROUP];   // exp(shifted logit), diag zeroed
    __shared__ float sL[GROUP];   // label distances for this row
    __shared__ float red[GROUP];  // reduction scratch
    int row = blockIdx.x;         // c*GROUP + i
    int i   = row & (GROUP - 1);
    int j   = threadIdx.x;
    size_t base = (size_t)row * GROUP;

    float lg = -0.5f * Dist[base + j];   // TEMP = 2.0
    sL[j]  = L[base + j];
    red[j] = lg;
    __syncthreads();
    for (int s = 128; s > 0; s >>= 1) {   // row max (includes diagonal, as in ref)
        if (j < s) red[j] = fmaxf(red[j], red[j + s]);
        __syncthreads();
    }
    float m = red[0];
    __syncthreads();

    float sl = lg - m;                    // shifted logit
    sE[j] = (j == i) ? 0.f : expf(sl);    // k == i excluded from denominator
    __syncthreads();

    float contrib = 0.f;
    if (j != i) {
        float Lj = sL[j];
        float denom = 0.f;
        for (int k = 0; k < GROUP; ++k)
            denom += (sL[k] >= Lj) ? sE[k] : 0.f;
        contrib = logf(denom) - sl;       // -log_prob
    }
    red[j] = contrib;
    __syncthreads();
    for (int s = 128; s > 0; s >>= 1) {
        if (j < s) red[j] += red[j + s];
        __syncthreads();
    }
    if (j == 0) rowsum[row] = red[0];
}

// ------------------------------------------------- deterministic final sum
__global__ void k_final(const float* __restrict__ rowsum, float* __restrict__ out) {
    __shared__ float red[256];
    int j = threadIdx.x;
    float s = 0.f;
    for (int t = j; t < NUM_CLASSES * GROUP; t += 256) s += rowsum[t];
    red[j] = s;
    __syncthreads();
    for (int st = 128; st > 0; st >>= 1) {
        if (j < st) red[j] += red[j + st];
        __syncthreads();
    }
    if (j == 0)
        out[0] = red[0] * (1.f / ((float)GROUP * (GROUP - 1) * NUM_CLASSES));
}

extern "C" void kernel_launch(void* const* d_in, const int* in_sizes, int n_in,
                              void* d_out, int out_size, void* d_ws, size_t ws_size,
                              hipStream_t stream) {
    (void)in_sizes; (void)n_in; (void)out_size; (void)ws_size;
    const float* F   = (const float*)d_in[0];   // features [1536,256]
    // d_in[1] = labels (unused; grouping is static contiguous)
    const float* g   = (const float*)d_in[2];   // gt_green [1536,3]
    const float* r   = (const float*)d_in[3];   // gt_red   [1536,3]
    const float* t   = (const float*)d_in[4];   // gt_trans [1536,3]
    const int*   sym = (const int*)d_in[5];     // sym      [1536,4]
    float* out = (float*)d_out;

    const int NN = NUM_CLASSES * GROUP * GROUP;           // 393216
    float* ws     = (float*)d_ws;
    float* Lws    = ws;                                   // label distances
    float* Dws    = Lws + NN;                             // feature distances
    float* n2     = Dws + NN;                             // squared norms (1536)
    float* rowsum = n2 + NUM_CLASSES * GROUP;             // per-row partials (1536)

    k_norms    <<<(NUM_CLASSES * GROUP) / 8, 256, 0, stream>>>(F, n2);
    k_labeldist<<<NN / 256,                  256, 0, stream>>>(g, r, t, sym, Lws);
    k_gram_dist<<<(NUM_CLASSES * 16 * 16) / 8, 256, 0, stream>>>(F, n2, Dws);
    k_rows     <<<NUM_CLASSES * GROUP,       256, 0, stream>>>(Dws, Lws, rowsum);
    k_final    <<<1,                         256, 0, stream>>>(rowsum, out);
}